// MashiDist_17317308137529
// MI455X (gfx1250) — compile-verified
//
#include <hip/hip_runtime.h>
#include <math.h>

typedef float v2f __attribute__((ext_vector_type(2)));
typedef float v8f __attribute__((ext_vector_type(8)));

// X = [x0; x1] is (512, 4096) row-major. 4096 points (columns), dim 512.
#define NPTS 4096
#define NDIM 512

__device__ __forceinline__ const float* xrow(const float* x0, const float* x1, int r) {
    return (r < 256) ? (x0 + (size_t)r * NPTS) : (x1 + (size_t)(r - 256) * NPTS);
}

// ---------------- K1: mu[i] = mean_k X[i,k] ----------------
__global__ void k_mean(const float* __restrict__ x0, const float* __restrict__ x1,
                       float* __restrict__ mu) {
    int row = blockIdx.x;
    const float* src = xrow(x0, x1, row);
    float s = 0.f;
    for (int k = threadIdx.x; k < NPTS; k += blockDim.x) s += src[k];
    __shared__ float red[256];
    red[threadIdx.x] = s; __syncthreads();
    for (int off = 128; off > 0; off >>= 1) {
        if (threadIdx.x < off) red[threadIdx.x] += red[threadIdx.x + off];
        __syncthreads();
    }
    if (threadIdx.x == 0) mu[row] = red[0] * (1.0f / (float)NPTS);
}

// ---------------- K2: V = Xc * Xc^T / (NPTS-1), 512x512, fp32 WMMA ----------------
// One 16x16 tile of V per wave; K = 4096 in steps of 4.
__global__ void k_cov(const float* __restrict__ x0, const float* __restrict__ x1,
                      const float* __restrict__ mu, float* __restrict__ V) {
    const int lane = threadIdx.x & 31;
    const int wave = threadIdx.x >> 5;
    const int tile = blockIdx.x * 4 + wave;       // 0..1023  (32x32 tile grid)
    const int ti = tile >> 5, tj = tile & 31;
    const int n15 = lane & 15, hi = lane >> 4;
    const int kk = hi * 2;                         // per-lane K sub-offset {0,2}
    const int rowA = ti * 16 + n15;                // A[m][k] = Xc[rowA][k], m = n15
    const int rowB = tj * 16 + n15;                // B[k][n] = Xc[rowB][k], n = n15
    const float* pa = xrow(x0, x1, rowA);
    const float* pb = xrow(x0, x1, rowB);
    const float muA = mu[rowA], muB = mu[rowB];
    v8f acc = {};
    for (int k0 = 0; k0 < NPTS; k0 += 4) {
        v2f a, b;
        a.x = pa[k0 + kk]     - muA;
        a.y = pa[k0 + kk + 1] - muA;
        b.x = pb[k0 + kk]     - muB;
        b.y = pb[k0 + kk + 1] - muB;
        acc = __builtin_amdgcn_wmma_f32_16x16x4_f32(false, a, false, b,
                                                    (short)0, acc, false, false);
    }
    const float scale = 1.0f / (float)(NPTS - 1);
    #pragma unroll
    for (int v = 0; v < 8; ++v) {
        int m = hi * 8 + v;                        // C/D layout: M = v + 8*hi, N = n15
        V[(size_t)(ti * 16 + m) * NDIM + tj * 16 + n15] = acc[v] * scale;
    }
}

// ---------------- K3: Cholesky V = L*L^T (single workgroup, in-place lower) ----------------
// Writes Lt[j][a] = L[a][j] (row j of Lt = column j of L, contiguous) and recip[j]=1/L[j][j].
__global__ void k_chol(float* __restrict__ V, float* __restrict__ Lt,
                       float* __restrict__ recip) {
    __shared__ float sh[2];
    __shared__ float colj[NDIM];
    const int tid = threadIdx.x;
    const int nt  = blockDim.x;     // 1024
    for (int j = 0; j < NDIM; ++j) {
        if (tid == 0) {
            float d = sqrtf(V[(size_t)j * NDIM + j]);
            sh[0] = d; sh[1] = 1.0f / d;
            recip[j] = 1.0f / d;
        }
        __syncthreads();
        float dval = sh[0], inv = sh[1];
        // scale column j (a >= j), stash in LDS, emit Lt row j
        for (int a = j + tid; a < NDIM; a += nt) {
            float val = (a == j) ? dval : V[(size_t)a * NDIM + j] * inv;
            V[(size_t)a * NDIM + j] = val;
            colj[a] = val;
            Lt[(size_t)j * NDIM + a] = val;
        }
        __syncthreads();
        // trailing lower-triangle rank-1 update: V[a][b] -= L[a][j]*L[b][j], j<b<=a
        for (int a = j + 1 + (tid >> 5); a < NDIM; a += (nt >> 5)) {
            float la = colj[a];
            float* rowa = V + (size_t)a * NDIM;
            for (int b = j + 1 + (tid & 31); b <= a; b += 32)
                rowa[b] -= la * colj[b];
        }
        __syncthreads();
    }
}

// ---------------- K4: Wt = (L^{-1} X)^T (forward substitution), 32 columns/block; fused q ----------------
// Wt is (NPTS x NDIM): Wt[point][k] — point-major so the Gram GEMM reads contiguous K.
__global__ void k_trsm(const float* __restrict__ x0, const float* __restrict__ x1,
                       const float* __restrict__ Lt, const float* __restrict__ recip,
                       float* __restrict__ Wt, float* __restrict__ q) {
    __shared__ float s[NDIM * 32];                 // 64 KB slab: s[i*32 + c]
    const int tid  = threadIdx.x;                  // 256
    const int col0 = blockIdx.x * 32;
    for (int idx = tid; idx < NDIM * 32; idx += 256) {
        int i = idx >> 5, c = idx & 31;
        s[idx] = xrow(x0, x1, i)[col0 + c];
    }
    __syncthreads();
    for (int i = 0; i < NDIM; ++i) {
        if (tid < 32) s[i * 32 + tid] *= recip[i];
        __syncthreads();
        const float* ltrow = Lt + (size_t)i * NDIM;   // column i of L
        int rem = NDIM - 1 - i;
        for (int idx = tid; idx < rem * 32; idx += 256) {
            int ip = i + 1 + (idx >> 5);
            int c  = idx & 31;
            s[ip * 32 + c] -= ltrow[ip] * s[i * 32 + c];
        }
        __syncthreads();
    }
    // transposed, coalesced write: consecutive threads -> consecutive k
    for (int idx = tid; idx < NDIM * 32; idx += 256) {
        int i = idx & (NDIM - 1);      // k index, consecutive per thread
        int c = idx >> 9;              // column within slab
        Wt[(size_t)(col0 + c) * NDIM + i] = s[i * 32 + c];
    }
    if (tid < 32) {
        float acc = 0.f;
        for (int i = 0; i < NDIM; ++i) { float w = s[i * 32 + tid]; acc += w * w; }
        q[col0 + tid] = acc;
    }
}

// ---------------- K5: G = Wt Wt^T (upper part) fused with condensed distance output ----------------
// 32x64 output per wave: 8 accumulators, 6 b64 operand loads -> 8 WMMAs per K-step of 4.
__device__ __forceinline__ void emit(float* __restrict__ out, int a, int b, float d2) {
    if (a < b) {
        long long idx = (long long)a * NPTS - ((long long)a * (a + 1)) / 2 + (b - a - 1);
        out[idx] = sqrtf(fmaxf(d2, 1e-12f));
    }
}

__global__ void k_gram_dist(const float* __restrict__ Wt, const float* __restrict__ q,
                            float* __restrict__ out) {
    const int lane = threadIdx.x & 31;
    const int wave = threadIdx.x >> 5;
    const int cbB = blockIdx.x * 4 + wave;        // 0..63 : b-cols [64*cbB, +64)
    const int raB = blockIdx.y;                   // 0..127: a-rows [32*raB, +32)
    if (raB >= 2 * cbB + 2) return;               // wave-uniform: block fully below diagonal
    const int n15 = lane & 15, hi = lane >> 4;
    const int kk = hi * 2;
    const int ca = raB * 32, cb = cbB * 64;
    const float* pa0 = Wt + (size_t)(ca + n15) * NDIM;
    const float* pa1 = pa0 + 16 * NDIM;
    const float* pb0 = Wt + (size_t)(cb + n15) * NDIM;
    const float* pb1 = pb0 + 16 * NDIM;
    const float* pb2 = pb0 + 32 * NDIM;
    const float* pb3 = pb0 + 48 * NDIM;
    v8f acc00 = {}, acc01 = {}, acc02 = {}, acc03 = {};
    v8f acc10 = {}, acc11 = {}, acc12 = {}, acc13 = {};
    for (int k0 = 0; k0 < NDIM; k0 += 4) {
        v2f a0, a1, b0, b1, b2, b3;
        a0.x = pa0[k0 + kk]; a0.y = pa0[k0 + kk + 1];
        a1.x = pa1[k0 + kk]; a1.y = pa1[k0 + kk + 1];
        b0.x = pb0[k0 + kk]; b0.y = pb0[k0 + kk + 1];
        b1.x = pb1[k0 + kk]; b1.y = pb1[k0 + kk + 1];
        b2.x = pb2[k0 + kk]; b2.y = pb2[k0 + kk + 1];
        b3.x = pb3[k0 + kk]; b3.y = pb3[k0 + kk + 1];
        acc00 = __builtin_amdgcn_wmma_f32_16x16x4_f32(false, a0, false, b0, (short)0, acc00, false, false);
        acc01 = __builtin_amdgcn_wmma_f32_16x16x4_f32(false, a0, false, b1, (short)0, acc01, false, false);
        acc02 = __builtin_amdgcn_wmma_f32_16x16x4_f32(false, a0, false, b2, (short)0, acc02, false, false);
        acc03 = __builtin_amdgcn_wmma_f32_16x16x4_f32(false, a0, false, b3, (short)0, acc03, false, false);
        acc10 = __builtin_amdgcn_wmma_f32_16x16x4_f32(false, a1, false, b0, (short)0, acc10, false, false);
        acc11 = __builtin_amdgcn_wmma_f32_16x16x4_f32(false, a1, false, b1, (short)0, acc11, false, false);
        acc12 = __builtin_amdgcn_wmma_f32_16x16x4_f32(false, a1, false, b2, (short)0, acc12, false, false);
        acc13 = __builtin_amdgcn_wmma_f32_16x16x4_f32(false, a1, false, b3, (short)0, acc13, false, false);
    }
    const float qb0 = q[cb + n15];
    const float qb1 = q[cb + 16 + n15];
    const float qb2 = q[cb + 32 + n15];
    const float qb3 = q[cb + 48 + n15];
    const int b0c = cb + n15, b1c = b0c + 16, b2c = b0c + 32, b3c = b0c + 48;
    #pragma unroll
    for (int v = 0; v < 8; ++v) {
        int a0r = ca + hi * 8 + v;
        int a1r = a0r + 16;
        float qa0 = q[a0r], qa1 = q[a1r];
        emit(out, a0r, b0c, qa0 + qb0 - 2.0f * acc00[v]);
        emit(out, a0r, b1c, qa0 + qb1 - 2.0f * acc01[v]);
        emit(out, a0r, b2c, qa0 + qb2 - 2.0f * acc02[v]);
        emit(out, a0r, b3c, qa0 + qb3 - 2.0f * acc03[v]);
        emit(out, a1r, b0c, qa1 + qb0 - 2.0f * acc10[v]);
        emit(out, a1r, b1c, qa1 + qb1 - 2.0f * acc11[v]);
        emit(out, a1r, b2c, qa1 + qb2 - 2.0f * acc12[v]);
        emit(out, a1r, b3c, qa1 + qb3 - 2.0f * acc13[v]);
    }
}

extern "C" void kernel_launch(void* const* d_in, const int* in_sizes, int n_in,
                              void* d_out, int out_size, void* d_ws, size_t ws_size,
                              hipStream_t stream) {
    (void)in_sizes; (void)n_in; (void)out_size; (void)ws_size;
    const float* x0 = (const float*)d_in[0];
    const float* x1 = (const float*)d_in[1];
    float* out = (float*)d_out;
    float* ws  = (float*)d_ws;
    // scratch layout (floats): mu[512] | recip[512] | q[4096] | V[512*512] | Lt[512*512] | Wt[4096*512]
    float* mu    = ws;
    float* recip = ws + 512;
    float* q     = ws + 1024;
    float* V     = ws + 5120;
    float* Lt    = V  + 512 * 512;
    float* Wt    = Lt + 512 * 512;

    k_mean<<<512, 256, 0, stream>>>(x0, x1, mu);
    k_cov <<<256, 128, 0, stream>>>(x0, x1, mu, V);
    k_chol<<<1, 1024, 0, stream>>>(V, Lt, recip);
    k_trsm<<<128, 256, 0, stream>>>(x0, x1, Lt, recip, Wt, q);
    k_gram_dist<<<dim3(16, 128), 128, 0, stream>>>(Wt, q, out);
}